// GNNML1_36721970380952
// MI455X (gfx1250) — compile-verified
//
#include <hip/hip_runtime.h>

#define N_NODES 50000
#define N_EDGES 800000
#define FEATS   128
#define HIDDEN  64

typedef __attribute__((ext_vector_type(2))) float v2f;
typedef __attribute__((ext_vector_type(4))) float v4f;
typedef __attribute__((ext_vector_type(8))) float v8f;

// ---------------------------------------------------------------------------
// Utility: zero a float buffer (n must be handled even if not /4; here it is)
// ---------------------------------------------------------------------------
__global__ void zero_f32(float* __restrict__ p, int n) {
    int i   = blockIdx.x * blockDim.x + threadIdx.x;
    int idx = i * 4;
    if (idx + 3 < n) {
        *(v4f*)(p + idx) = (v4f){0.f, 0.f, 0.f, 0.f};
    } else if (idx < n) {
        for (int j = idx; j < n; ++j) p[j] = 0.f;
    }
}

// ---------------------------------------------------------------------------
// Degree accumulation + inverse-sqrt normalization
// ---------------------------------------------------------------------------
__global__ void deg_accum(const int* __restrict__ dst, float* __restrict__ deg) {
    int e = blockIdx.x * blockDim.x + threadIdx.x;
    if (e < N_EDGES) atomicAdd(&deg[dst[e]], 1.0f);
}

__global__ void inv_sqrt_k(float* __restrict__ deg) {
    int i = blockIdx.x * blockDim.x + threadIdx.x;
    if (i < N_NODES) deg[i] = rsqrtf(fmaxf(deg[i], 1.0f));
}

// ---------------------------------------------------------------------------
// Edge scatter: conv[dst] += h[src] * inv_sqrt[src]*inv_sqrt[dst]
// One thread handles one edge x 4 features (float4 gather, 4 atomic adds).
// ---------------------------------------------------------------------------
__global__ void edge_conv(const int* __restrict__ src, const int* __restrict__ dst,
                          const float* __restrict__ insq,
                          const float* __restrict__ h, float* __restrict__ out) {
    long tid = (long)blockIdx.x * blockDim.x + threadIdx.x;
    const long total = (long)N_EDGES * (HIDDEN / 4);
    if (tid >= total) return;
    int e = (int)(tid >> 4);          // HIDDEN/4 == 16 slots per edge
    int f = ((int)tid & 15) << 2;     // feature group of 4
    int s = src[e], d = dst[e];
    float c = insq[s] * insq[d];
    v4f v = *(const v4f*)(h + (size_t)s * HIDDEN + f);
    float* o = out + (size_t)d * HIDDEN + f;
    atomicAdd(o + 0, v.x * c);
    atomicAdd(o + 1, v.y * c);
    atomicAdd(o + 2, v.z * c);
    atomicAdd(o + 3, v.w * c);
}

// ---------------------------------------------------------------------------
// Layer-1 GEMM: out = relu(X[M,K] @ W[N,K]^T + bias), M=50000, N=64.
// One wave32 per 16x16 tile via V_WMMA_F32_16X16X4_F32.
// blockDim = 128 (4 waves -> all 64 output cols), gridDim = M/16 = 3125.
// ---------------------------------------------------------------------------
__global__ void gemm_relu_bias(const float* __restrict__ X, const float* __restrict__ W,
                               const float* __restrict__ bias, float* __restrict__ out,
                               int K) {
    const int lane  = threadIdx.x & 31;
    const int wave  = threadIdx.x >> 5;
    const int half  = lane >> 4;    // 0: K={0,1}, 1: K={2,3} within each k-step
    const int l16   = lane & 15;
    const int tileM = blockIdx.x * 16;
    const int tileN = wave * 16;

    const float* xrow = X + (size_t)(tileM + l16) * K + 2 * half;  // A: M on lanes
    const float* wrow = W + (size_t)(tileN + l16) * K + 2 * half;  // B: N on lanes

    v8f acc = {};
    for (int k = 0; k < K; k += 4) {
        v2f a = *(const v2f*)(xrow + k);
        v2f b = *(const v2f*)(wrow + k);
        acc = __builtin_amdgcn_wmma_f32_16x16x4_f32(false, a, false, b,
                                                    (short)0, acc, false, false);
    }

    float bv = bias[tileN + l16];
    float* obase = out + (size_t)(tileM + 8 * half) * HIDDEN + tileN + l16;
#pragma unroll
    for (int v = 0; v < 8; ++v)
        obase[(size_t)v * HIDDEN] = fmaxf(acc[v] + bv, 0.0f);
}

// ---------------------------------------------------------------------------
// Fused block: out = relu( [x, conv] @ W^T  +  (x @ Aa^T) * (x @ Ab^T) )
// x, conv: [N_NODES, 64]; W: [64, 128]; Aa, Ab: [64, 64].
// Shared A-fragments (x) feed three accumulators. All K loops fp32 WMMA.
// ---------------------------------------------------------------------------
__global__ void block_fused(const float* __restrict__ X, const float* __restrict__ CONV,
                            const float* __restrict__ W, const float* __restrict__ Aa,
                            const float* __restrict__ Ab, float* __restrict__ out) {
    const int lane  = threadIdx.x & 31;
    const int wave  = threadIdx.x >> 5;
    const int half  = lane >> 4;
    const int l16   = lane & 15;
    const int tileM = blockIdx.x * 16;
    const int tileN = wave * 16;

    const float* xrow = X    + (size_t)(tileM + l16) * HIDDEN + 2 * half;
    const float* crow = CONV + (size_t)(tileM + l16) * HIDDEN + 2 * half;
    const float* w0   = W  + (size_t)(tileN + l16) * (2 * HIDDEN) + 2 * half; // K 0..63 (x)
    const float* w1   = w0 + HIDDEN;                                          // K 64..127 (conv)
    const float* aar  = Aa + (size_t)(tileN + l16) * HIDDEN + 2 * half;
    const float* abr  = Ab + (size_t)(tileN + l16) * HIDDEN + 2 * half;

    v8f accW = {}, accA = {}, accB = {};
    for (int k = 0; k < HIDDEN; k += 4) {
        v2f ax  = *(const v2f*)(xrow + k);
        v2f ac  = *(const v2f*)(crow + k);
        v2f bw0 = *(const v2f*)(w0 + k);
        v2f bw1 = *(const v2f*)(w1 + k);
        v2f ba  = *(const v2f*)(aar + k);
        v2f bb  = *(const v2f*)(abr + k);
        accW = __builtin_amdgcn_wmma_f32_16x16x4_f32(false, ax, false, bw0,
                                                     (short)0, accW, false, false);
        accW = __builtin_amdgcn_wmma_f32_16x16x4_f32(false, ac, false, bw1,
                                                     (short)0, accW, false, false);
        accA = __builtin_amdgcn_wmma_f32_16x16x4_f32(false, ax, false, ba,
                                                     (short)0, accA, false, false);
        accB = __builtin_amdgcn_wmma_f32_16x16x4_f32(false, ax, false, bb,
                                                     (short)0, accB, false, false);
    }

    float* obase = out + (size_t)(tileM + 8 * half) * HIDDEN + tileN + l16;
#pragma unroll
    for (int v = 0; v < 8; ++v)
        obase[(size_t)v * HIDDEN] = fmaxf(accW[v] + accA[v] * accB[v], 0.0f);
}

// ---------------------------------------------------------------------------
// Host launcher
// ---------------------------------------------------------------------------
extern "C" void kernel_launch(void* const* d_in, const int* in_sizes, int n_in,
                              void* d_out, int out_size, void* d_ws, size_t ws_size,
                              hipStream_t stream) {
    const float* x   = (const float*)d_in[0];
    const int* edges = (const int*)d_in[1];
    const int* src   = edges;
    const int* dst   = edges + N_EDGES;
    const float* W1  = (const float*)d_in[2];
    const float* b1  = (const float*)d_in[3];
    const float* W2  = (const float*)d_in[4];
    const float* A2a = (const float*)d_in[5];
    const float* A2b = (const float*)d_in[6];
    const float* W3  = (const float*)d_in[7];
    const float* A3a = (const float*)d_in[8];
    const float* A3b = (const float*)d_in[9];
    const float* W4  = (const float*)d_in[10];
    const float* A4a = (const float*)d_in[11];
    const float* A4b = (const float*)d_in[12];
    float* outp = (float*)d_out;

    // Workspace partition
    char* ws = (char*)d_ws;
    float* deg = (float*)ws;
    size_t off = ((size_t)N_NODES * 4 + 255) & ~(size_t)255;
    float* bufA = (float*)(ws + off); off += (size_t)N_NODES * HIDDEN * 4;
    float* bufC = (float*)(ws + off); off += (size_t)N_NODES * HIDDEN * 4;
    float* bufB = (float*)(ws + off);

    const int T = 256;
    const int NH = N_NODES * HIDDEN;          // 3,200,000

    // Degrees -> inverse sqrt
    zero_f32<<<((N_NODES + 3) / 4 + T - 1) / T, T, 0, stream>>>(deg, N_NODES);
    deg_accum<<<(N_EDGES + T - 1) / T, T, 0, stream>>>(dst, deg);
    inv_sqrt_k<<<(N_NODES + T - 1) / T, T, 0, stream>>>(deg);

    // Layer 1: h0 = relu(x @ W1^T + b1)
    gemm_relu_bias<<<N_NODES / 16, 128, 0, stream>>>(x, W1, b1, bufA, FEATS);

    const long econv_threads = (long)N_EDGES * (HIDDEN / 4);
    const int  econv_blocks  = (int)((econv_threads + T - 1) / T);

    // Block 2: bufA -> bufB
    zero_f32<<<(NH / 4 + T - 1) / T, T, 0, stream>>>(bufC, NH);
    edge_conv<<<econv_blocks, T, 0, stream>>>(src, dst, deg, bufA, bufC);
    block_fused<<<N_NODES / 16, 128, 0, stream>>>(bufA, bufC, W2, A2a, A2b, bufB);

    // Block 3: bufB -> bufA
    zero_f32<<<(NH / 4 + T - 1) / T, T, 0, stream>>>(bufC, NH);
    edge_conv<<<econv_blocks, T, 0, stream>>>(src, dst, deg, bufB, bufC);
    block_fused<<<N_NODES / 16, 128, 0, stream>>>(bufB, bufC, W3, A3a, A3b, bufA);

    // Block 4: bufA -> d_out
    zero_f32<<<(NH / 4 + T - 1) / T, T, 0, stream>>>(bufC, NH);
    edge_conv<<<econv_blocks, T, 0, stream>>>(src, dst, deg, bufA, bufC);
    block_fused<<<N_NODES / 16, 128, 0, stream>>>(bufA, bufC, W4, A4a, A4b, outp);
}